// Codebook_33535104647888
// MI455X (gfx1250) — compile-verified
//
#include <hip/hip_runtime.h>

#define B_   8
#define R_   2048
#define IN_  512
#define H_   4
#define K_   4096
#define D_   256
#define N_   (B_ * R_)    /* 16384 */
#define HD_  (H_ * D_)    /* 1024  */
#define BETA 0.25f

typedef __attribute__((ext_vector_type(2))) float v2f;
typedef __attribute__((ext_vector_type(4))) float v4f;
typedef __attribute__((ext_vector_type(8))) float v8f;

// ---- workspace layout (in floats) ----
#define WS_ZH   ((size_t)0)                        // H*N*D  = 16,777,216 f
#define WS_CBT  (WS_ZH  + (size_t)H_ * N_ * D_)    // H*(D/2)*K*2 = 4,194,304 f
#define WS_CE   (WS_CBT + (size_t)H_ * D_ * K_)    // H*K    =     16,384 f
#define WS_IDX  (WS_CE  + (size_t)H_ * K_)         // H*N    =     65,536 i32

__device__ __forceinline__ v8f wmma_f32(v2f a, v2f b, v8f c) {
  // D = A(16x4,f32) * B(4x16,f32) + C(16x16,f32)  -> v_wmma_f32_16x16x4_f32
  return __builtin_amdgcn_wmma_f32_16x16x4_f32(
      /*neg_a=*/false, a, /*neg_b=*/false, b,
      /*c_mod=*/(short)0, c, /*reuse_a=*/false, /*reuse_b=*/false);
}

// CDNA5 async memory->LDS copy (ASYNCcnt-tracked, no VGPR data movement).
__device__ __forceinline__ void async_ld_b128(unsigned lds_addr, const void* gptr) {
  asm volatile("global_load_async_to_lds_b128 %0, %1, off"
               :: "v"(lds_addr), "v"(gptr) : "memory");
}
__device__ __forceinline__ void wait_async0() {
  asm volatile("s_wait_asynccnt 0x0" ::: "memory");
}
__device__ __forceinline__ unsigned lds_addr_of(const void* p) {
  return (unsigned)(size_t)p;   // shared-aperture generic addr: low 32 bits = LDS offset
}

// ---------------------------------------------------------------------------
// codebook (H,K,D) -> d-pair-interleaved transpose cbTp[h][d/2][k][2]
// so a distance-GEMM B-fragment (values at d=kk,kk+1, fixed k) is one b64 load
// and 16 lanes form one contiguous 128B request.
// ---------------------------------------------------------------------------
__global__ __launch_bounds__(256) void k_transpose(const float* __restrict__ cb,
                                                   float* __restrict__ cbTp) {
  __shared__ float t[32][33];
  const int h  = blockIdx.z;
  const int k0 = blockIdx.x * 32;
  const int d0 = blockIdx.y * 32;
  const int tx = threadIdx.x & 31;
  const int ty = threadIdx.x >> 5;            // 0..7
  for (int i = 0; i < 32; i += 8)
    t[ty + i][tx] = cb[((size_t)h * K_ + (k0 + ty + i)) * D_ + d0 + tx];
  __syncthreads();
  for (int i = 0; i < 32; i += 8) {
    const int d = d0 + ty + i;
    cbTp[(((size_t)h * (D_/2) + (d >> 1)) * K_ + (k0 + tx)) * 2 + (d & 1)] = t[tx][ty + i];
  }
}

// ---------------------------------------------------------------------------
// ce[h*K + k] = sum_d codebook[h,k,d]^2
// ---------------------------------------------------------------------------
__global__ __launch_bounds__(256) void k_ce(const float* __restrict__ cb,
                                            float* __restrict__ ce) {
  const int i = blockIdx.x * 256 + threadIdx.x;    // flat (h,k)
  const float* row = cb + (size_t)i * D_;
  float s = 0.f;
  for (int d = 0; d < D_; d += 4) {
    v4f v = *(const v4f*)(row + d);
    s += v.x * v.x + v.y * v.y + v.z * v.z + v.w * v.w;
  }
  ce[i] = s;
}

// ---------------------------------------------------------------------------
// zh = reshape(z @ Wq + bq)  stored as (H, N, D)
// 64x64 tile per block; A staged via async-to-LDS, B staged transposed so the
// WMMA B-fragment is a single b64 LDS read.
// ---------------------------------------------------------------------------
__global__ __launch_bounds__(256) void k_gemm_za(const float* __restrict__ z,
                                                 const float* __restrict__ Wq,
                                                 const float* __restrict__ bq,
                                                 float* __restrict__ zh) {
  __shared__ float As[64][20];   // [row][kk], rows 80B (16B aligned), bank-stride 20
  __shared__ float Bs[64][20];   // [col][kk], transposed
  const int tid  = threadIdx.x;
  const int m0   = blockIdx.x * 64;
  const int n0   = blockIdx.y * 64;
  const int wave = tid >> 5, lane = tid & 31;
  const int h16  = lane >> 4, l16 = lane & 15;
  const int wr   = wave >> 1;          // subtile row 0..3
  const int wc0  = (wave & 1) * 2;     // subtile cols wc0, wc0+1

  const int ar = tid >> 2, ac = (tid & 3) * 4;     // A staging coords
  const int br = tid >> 4, bc = (tid & 15) * 4;    // B staging coords
  const unsigned a_lds = lds_addr_of(&As[ar][ac]);

  v8f acc0 = {0.f,0.f,0.f,0.f,0.f,0.f,0.f,0.f};
  v8f acc1 = {0.f,0.f,0.f,0.f,0.f,0.f,0.f,0.f};

  for (int k0 = 0; k0 < IN_; k0 += 16) {
    // A tile 64x16: async global->LDS, 16B per lane
    async_ld_b128(a_lds, z + (size_t)(m0 + ar) * IN_ + k0 + ac);
    // B tile 16x64, stored transposed Bs[col][k]
    {
      v4f v = *(const v4f*)(Wq + (size_t)(k0 + br) * HD_ + n0 + bc);
      Bs[bc  ][br] = v.x; Bs[bc+1][br] = v.y; Bs[bc+2][br] = v.z; Bs[bc+3][br] = v.w;
    }
    if (k0 + 16 < IN_)
      __builtin_prefetch(Wq + (size_t)(k0 + 16 + br) * HD_ + n0 + bc, 0, 1);
    wait_async0();
    __syncthreads();

    const int arow = wr * 16 + l16;
    for (int d0 = 0; d0 < 16; d0 += 4) {
      const int kk = d0 + 2 * h16;
      v2f a  = *(const v2f*)&As[arow][kk];
      v2f b0 = *(const v2f*)&Bs[wc0 * 16 + l16][kk];
      acc0 = wmma_f32(a, b0, acc0);
      v2f b1 = *(const v2f*)&Bs[(wc0 + 1) * 16 + l16][kk];
      acc1 = wmma_f32(a, b1, acc1);
    }
    __syncthreads();
  }
  // bias + scatter into (H, N, D) layout
  for (int t = 0; t < 2; t++) {
    const v8f acc = t ? acc1 : acc0;
    const int col = n0 + (wc0 + t) * 16 + l16;
    const float bias = bq[col];
    const int h = col >> 8, d = col & 255;
    for (int j = 0; j < 8; j++) {
      const int row = m0 + wr * 16 + j + 8 * h16;
      zh[((size_t)h * N_ + row) * D_ + d] = acc[j] + bias;
    }
  }
}

// ---------------------------------------------------------------------------
// argmin_k ( ce[k] - 2 * zh . e_k )   per (h, n)
// zh tile async-staged to LDS; B-fragments are coalesced b64 global loads from
// the d-pair-interleaved codebook. f32 WMMA (argmin gaps ~5e-7 forbid bf16).
// ---------------------------------------------------------------------------
__global__ __launch_bounds__(256) void k_argmin(const float* __restrict__ zh,
                                                const float* __restrict__ cbTp,
                                                const float* __restrict__ ce,
                                                int*   __restrict__ idx_ws,
                                                float* __restrict__ out_idx) {
  __shared__ float As[16][260];   // rows 1040B: 16B-aligned, bank-stride 4
  __shared__ float rv[8][16];
  __shared__ int   ri[8][16];
  const int m0  = blockIdx.x * 16;
  const int h   = blockIdx.y;
  const int tid = threadIdx.x;

  const float* zbase = zh + ((size_t)h * N_ + m0) * D_;
  for (int i = tid; i < 16 * D_ / 4; i += 256) {   // 1024 float4 via async
    const int row = i >> 6, c4 = (i & 63) * 4;
    async_ld_b128(lds_addr_of(&As[row][c4]), zbase + (size_t)row * D_ + c4);
  }
  wait_async0();
  __syncthreads();

  const int wave = tid >> 5, lane = tid & 31;
  const int h16 = lane >> 4, l16 = lane & 15;
  const float* bb     = cbTp + (size_t)h * (D_/2) * K_ * 2;
  const float* cebase = ce + (size_t)h * K_;

  float minv[8]; int mini[8];
  for (int j = 0; j < 8; j++) { minv[j] = 3.4028235e38f; mini[j] = 0; }

  for (int kb = 0; kb < K_; kb += 128) {
    const int kc = kb + wave * 16;
    v8f acc = {0.f,0.f,0.f,0.f,0.f,0.f,0.f,0.f};
    for (int d0 = 0; d0 < D_; d0 += 4) {
      const int kk = d0 + 2 * h16;               // even
      v2f a = *(const v2f*)&As[l16][kk];
      v2f b = *(const v2f*)(bb + ((size_t)(kk >> 1) * K_ + kc + l16) * 2);
      acc = wmma_f32(a, b, acc);
    }
    const float cev = cebase[kc + l16];
    const int   kg  = kc + l16;
    for (int j = 0; j < 8; j++) {
      const float s = cev - 2.f * acc[j];
      if (s < minv[j]) { minv[j] = s; mini[j] = kg; }
    }
  }
  // reduce across the 16 lanes of each half (xor masks stay within half)
  for (int m = 1; m < 16; m <<= 1) {
    for (int j = 0; j < 8; j++) {
      const float ov = __shfl_xor(minv[j], m, 32);
      const int   oi = __shfl_xor(mini[j], m, 32);
      if (ov < minv[j] || (ov == minv[j] && oi < mini[j])) { minv[j] = ov; mini[j] = oi; }
    }
  }
  if (l16 == 0) {
    for (int j = 0; j < 8; j++) { rv[wave][j + 8*h16] = minv[j]; ri[wave][j + 8*h16] = mini[j]; }
  }
  __syncthreads();
  if (tid < 16) {
    float bv = rv[0][tid]; int bi = ri[0][tid];
    for (int w = 1; w < 8; w++) {
      const float v = rv[w][tid]; const int i = ri[w][tid];
      if (v < bv || (v == bv && i < bi)) { bv = v; bi = i; }
    }
    idx_ws[(size_t)h * N_ + m0 + tid]  = bi;
    out_idx[(size_t)h * N_ + m0 + tid] = (float)bi;
  }
}

// ---------------------------------------------------------------------------
// loss[n] = (1+BETA)/(H*D) * sum_{h,d} (cb[h,idx,d] - zh[h,n,d])^2
// (enc and code terms are numerically identical; sg only affects gradients)
// ---------------------------------------------------------------------------
__global__ __launch_bounds__(256) void k_loss(const float* __restrict__ zh,
                                              const float* __restrict__ cb,
                                              const int*   __restrict__ idx_ws,
                                              float* __restrict__ out_loss) {
  const int wave = threadIdx.x >> 5, lane = threadIdx.x & 31;
  const int n = blockIdx.x * 8 + wave;
  float s = 0.f;
  for (int h = 0; h < H_; h++) {
    const int id = idx_ws[(size_t)h * N_ + n];
    const float* zr = zh + ((size_t)h * N_ + n) * D_;
    const float* cr = cb + ((size_t)h * K_ + id) * D_;
    for (int d = lane; d < D_; d += 32) {
      const float df = cr[d] - zr[d];
      s += df * df;
    }
  }
  for (int m = 16; m >= 1; m >>= 1) s += __shfl_xor(s, m, 32);
  if (lane == 0) out_loss[n] = s * ((1.f + BETA) / (float)(H_ * D_));
}

// ---------------------------------------------------------------------------
// out = gather(codebook, idx) @ Wp + bp     (zq_ste == zq numerically)
// A-tile gathered on the fly through idx using per-lane async-to-LDS loads.
// ---------------------------------------------------------------------------
__global__ __launch_bounds__(256) void k_gemm_out(const float* __restrict__ cb,
                                                  const int*   __restrict__ idx_ws,
                                                  const float* __restrict__ Wp,
                                                  const float* __restrict__ bp,
                                                  float* __restrict__ out) {
  __shared__ float As[64][20];
  __shared__ float Bs[64][20];   // transposed [col][kk]
  const int tid  = threadIdx.x;
  const int m0   = blockIdx.x * 64;
  const int n0   = blockIdx.y * 64;
  const int wave = tid >> 5, lane = tid & 31;
  const int h16  = lane >> 4, l16 = lane & 15;
  const int wr   = wave >> 1;
  const int wc0  = (wave & 1) * 2;

  const int ar = tid >> 2, ac = (tid & 3) * 4;
  const int br = tid >> 4, bc = (tid & 15) * 4;
  const unsigned a_lds = lds_addr_of(&As[ar][ac]);

  v8f acc0 = {0.f,0.f,0.f,0.f,0.f,0.f,0.f,0.f};
  v8f acc1 = {0.f,0.f,0.f,0.f,0.f,0.f,0.f,0.f};

  for (int k0 = 0; k0 < HD_; k0 += 16) {
    const int h = k0 >> 8, dbase = k0 & 255;     // each 16-wide k-tile is inside one head
    {   // gathered A tile: per-lane address -> async copy straight into LDS
      const int id = idx_ws[(size_t)h * N_ + m0 + ar];
      async_ld_b128(a_lds, cb + ((size_t)h * K_ + id) * D_ + dbase + ac);
    }
    {   // B tile from Wp, stored transposed
      v4f v = *(const v4f*)(Wp + (size_t)(k0 + br) * IN_ + n0 + bc);
      Bs[bc  ][br] = v.x; Bs[bc+1][br] = v.y; Bs[bc+2][br] = v.z; Bs[bc+3][br] = v.w;
    }
    if (k0 + 16 < HD_)
      __builtin_prefetch(Wp + (size_t)(k0 + 16 + br) * IN_ + n0 + bc, 0, 1);
    wait_async0();
    __syncthreads();

    const int arow = wr * 16 + l16;
    for (int d0 = 0; d0 < 16; d0 += 4) {
      const int kk = d0 + 2 * h16;
      v2f a  = *(const v2f*)&As[arow][kk];
      v2f b0 = *(const v2f*)&Bs[wc0 * 16 + l16][kk];
      acc0 = wmma_f32(a, b0, acc0);
      v2f b1 = *(const v2f*)&Bs[(wc0 + 1) * 16 + l16][kk];
      acc1 = wmma_f32(a, b1, acc1);
    }
    __syncthreads();
  }
  for (int t = 0; t < 2; t++) {
    const v8f acc = t ? acc1 : acc0;
    const int col = n0 + (wc0 + t) * 16 + l16;
    const float bias = bp[col];
    for (int j = 0; j < 8; j++) {
      const int row = m0 + wr * 16 + j + 8 * h16;
      out[(size_t)row * IN_ + col] = acc[j] + bias;
    }
  }
}

// ---------------------------------------------------------------------------
extern "C" void kernel_launch(void* const* d_in, const int* in_sizes, int n_in,
                              void* d_out, int out_size, void* d_ws, size_t ws_size,
                              hipStream_t stream) {
  const float* z  = (const float*)d_in[0];
  const float* Wq = (const float*)d_in[1];
  const float* bq = (const float*)d_in[2];
  const float* cb = (const float*)d_in[3];
  const float* Wp = (const float*)d_in[4];
  const float* bp = (const float*)d_in[5];

  float* out      = (float*)d_out;
  float* out_idx  = out + (size_t)N_ * IN_;           // (H, B, R) as float
  float* out_loss = out_idx + (size_t)H_ * N_;        // (B, R)

  float* ws   = (float*)d_ws;
  float* zh   = ws + WS_ZH;
  float* cbTp = ws + WS_CBT;
  float* ce   = ws + WS_CE;
  int*   idx  = (int*)(ws + WS_IDX);

  k_transpose<<<dim3(K_ / 32, D_ / 32, H_), 256, 0, stream>>>(cb, cbTp);
  k_ce       <<<dim3(H_ * K_ / 256),        256, 0, stream>>>(cb, ce);
  k_gemm_za  <<<dim3(N_ / 64, HD_ / 64),    256, 0, stream>>>(z, Wq, bq, zh);
  k_argmin   <<<dim3(N_ / 16, H_),          256, 0, stream>>>(zh, cbTp, ce, idx, out_idx);
  k_loss     <<<dim3(N_ / 8),               256, 0, stream>>>(zh, cb, idx, out_loss);
  k_gemm_out <<<dim3(N_ / 64, IN_ / 64),    256, 0, stream>>>(cb, idx, Wp, bp, out);
}